// SGHN_71880572666047
// MI455X (gfx1250) — compile-verified
//
#include <hip/hip_runtime.h>
#include <hip/hip_bf16.h>

#define NP 10000
#define NE 160000
#define NB 16
#define H2 256
#define KSPLIT 25            // K4: split K=10000 into 25 chunks of 400
#define KCHUNK (NP / KSPLIT) // 400, divisible by 8

typedef float v2f __attribute__((ext_vector_type(2)));
typedef float v8f __attribute__((ext_vector_type(8)));

struct MLPp { const float *W1, *b1, *W2, *b2; };
struct Params {
  MLPp node_up, edge_init, edge_up1, edge_up2, net2, V1, V2, V3, V, T;
};

// ---- small-MLP helpers (x @ W1 -> tanh -> @ W2) --------------------------
template<int DIN, int DH>
__device__ __forceinline__ void mlp_hidden(const float* in, const float* W1,
                                           const float* b1, float* h) {
  #pragma unroll
  for (int k = 0; k < DH; ++k) {
    float acc = b1[k];
    #pragma unroll
    for (int i = 0; i < DIN; ++i) acc += in[i] * W1[i * DH + k];
    h[k] = tanhf(acc);
  }
}
template<int DH, int DOUT>
__device__ __forceinline__ void mlp_out(const float* h, const float* W2,
                                        const float* b2, float* out) {
  #pragma unroll
  for (int j = 0; j < DOUT; ++j) {
    float acc = b2[j];
    #pragma unroll
    for (int k = 0; k < DH; ++k) acc += h[k] * W2[k * DOUT + j];
    out[j] = acc;
  }
}
// accumulate d(loss)/d(input) given upstream grad and post-tanh hidden h
template<int DIN, int DH, int DOUT>
__device__ __forceinline__ void mlp_bwd_in(const float* g_out, const float* h,
                                           const float* W1, const float* W2,
                                           float* g_in) {
  float g_pre[DH];
  #pragma unroll
  for (int k = 0; k < DH; ++k) {
    float gh = 0.f;
    #pragma unroll
    for (int j = 0; j < DOUT; ++j) gh += g_out[j] * W2[k * DOUT + j];
    g_pre[k] = gh * (1.f - h[k] * h[k]);
  }
  #pragma unroll
  for (int i = 0; i < DIN; ++i) {
    float acc = 0.f;
    #pragma unroll
    for (int k = 0; k < DH; ++k) acc += g_pre[k] * W1[i * DH + k];
    g_in[i] += acc;
  }
}

// ---- K0: zero scratch accumulators ---------------------------------------
__global__ void kzero(float* p, int n) {
  int i = blockIdx.x * 256 + threadIdx.x;
  if (i < n) p[i] = 0.f;
}

// ---- K1: edge forward pass 1: agg[b, s[e]] += node_up(edge_init(eij)) ----
__global__ void k1_edge_fwd1(const float* __restrict__ x, const int* __restrict__ ei,
                             Params P, float* __restrict__ agg) {
  int idx = blockIdx.x * 256 + threadIdx.x;   // B*E exact
  int b = idx / NE, e = idx % NE;
  int s = ei[e], r = ei[NE + e];
  const float* q = x + b * 2 * NP;
  float eij = q[s] - q[r];
  float hei[5]; mlp_hidden<1, 5>(&eij, P.edge_init.W1, P.edge_init.b1, hei);
  float ee0[5]; mlp_out<5, 5>(hei, P.edge_init.W2, P.edge_init.b2, ee0);
  float hnu[5]; mlp_hidden<5, 5>(ee0, P.node_up.W1, P.node_up.b1, hnu);
  float nu[5];  mlp_out<5, 5>(hnu, P.node_up.W2, P.node_up.b2, nu);
  float* a = agg + (b * NP + s) * 5;
  #pragma unroll
  for (int i = 0; i < 5; ++i) atomicAdd(&a[i], nu[i]);
}

// ---- K2: edge forward pass 2: vv[b, s[e]] += V2(ee) ----------------------
__global__ void k2_edge_fwd2(const float* __restrict__ x, const int* __restrict__ ei,
                             Params P, const float* __restrict__ agg,
                             float* __restrict__ vv) {
  int idx = blockIdx.x * 256 + threadIdx.x;
  int b = idx / NE, e = idx % NE;
  int s = ei[e], r = ei[NE + e];
  const float* q = x + b * 2 * NP;
  float eij = q[s] - q[r];
  float hei[5]; mlp_hidden<1, 5>(&eij, P.edge_init.W1, P.edge_init.b1, hei);
  float ee0[5]; mlp_out<5, 5>(hei, P.edge_init.W2, P.edge_init.b2, ee0);
  const float* aS = agg + (b * NP + s) * 5;
  const float* aR = agg + (b * NP + r) * 5;
  float node1[5];
  #pragma unroll
  for (int i = 0; i < 5; ++i) node1[i] = aS[i] * aR[i];
  float h1[5]; mlp_hidden<5, 5>(node1, P.edge_up1.W1, P.edge_up1.b1, h1);
  float n11[5]; mlp_out<5, 5>(h1, P.edge_up1.W2, P.edge_up1.b2, n11);
  float h2[5]; mlp_hidden<5, 5>(node1, P.edge_up2.W1, P.edge_up2.b1, h2);
  float n12[5]; mlp_out<5, 5>(h2, P.edge_up2.W2, P.edge_up2.b2, n12);
  float ee[5];
  #pragma unroll
  for (int i = 0; i < 5; ++i) ee[i] = ee0[i] + 0.5f * (n11[i] + n12[i]);
  float hv2[5]; mlp_hidden<5, 5>(ee, P.V2.W1, P.V2.b1, hv2);
  float v2v[1]; mlp_out<5, 1>(hv2, P.V2.W2, P.V2.b2, v2v);
  atomicAdd(&vv[b * NP + s], v2v[0]);
}

// ---- K3: node forward: h11 = V([V1(agg), vv, V3(q)]) ---------------------
__global__ void k3_node_fwd(const float* __restrict__ x, Params P,
                            const float* __restrict__ agg, const float* __restrict__ vv,
                            float* __restrict__ h11) {
  int idx = blockIdx.x * 256 + threadIdx.x;   // B*NP exact
  int b = idx / NP, n = idx % NP;
  const float* a = agg + (b * NP + n) * 5;
  float hv1[5]; mlp_hidden<5, 5>(a, P.V1.W1, P.V1.b1, hv1);
  float v1v[1]; mlp_out<5, 1>(hv1, P.V1.W2, P.V1.b2, v1v);
  float qv = x[b * 2 * NP + n];
  float hv3[5]; mlp_hidden<1, 5>(&qv, P.V3.W1, P.V3.b1, hv3);
  float v3v[1]; mlp_out<5, 1>(hv3, P.V3.W2, P.V3.b2, v3v);
  float u[3] = {v1v[0], vv[b * NP + n], v3v[0]};
  float hv[5]; mlp_hidden<3, 5>(u, P.V.W1, P.V.b1, hv);
  float o[1]; mlp_out<5, 1>(hv, P.V.W2, P.V.b2, o);
  h11[b * NP + n] = o[0];
}

// ---- K4 (WMMA): partial y = h11 @ W1net2 over one K-chunk ----------------
// grid = 16 N-tiles x KSPLIT K-chunks; two independent accumulators so the
// hardware can pipeline independent WMMA chains; partials atomically summed
// into the zero-initialized y accumulator.
__global__ void k4_net2_fwd(const float* __restrict__ h11, Params P,
                            float* __restrict__ y_acc) {
  const int lane = threadIdx.x;
  const int half = lane >> 4, m = lane & 15;
  const int n0 = (blockIdx.x & 15) * 16;
  const int k0 = (blockIdx.x >> 4) * KCHUNK;
  const float* __restrict__ W1 = P.net2.W1;   // (10000, 256) row-major
  v8f c0 = {}, c1 = {};
  for (int k = k0; k < k0 + KCHUNK; k += 8) {
    const int kk = k + 2 * half;
    v2f a0, b0, a1, b1;
    a0.x = h11[m * NP + kk];
    a0.y = h11[m * NP + kk + 1];
    b0.x = W1[kk * H2 + n0 + m];
    b0.y = W1[(kk + 1) * H2 + n0 + m];
    const int k2 = kk + 4;
    a1.x = h11[m * NP + k2];
    a1.y = h11[m * NP + k2 + 1];
    b1.x = W1[k2 * H2 + n0 + m];
    b1.y = W1[(k2 + 1) * H2 + n0 + m];
    c0 = __builtin_amdgcn_wmma_f32_16x16x4_f32(false, a0, false, b0,
                                               (short)0, c0, false, false);
    c1 = __builtin_amdgcn_wmma_f32_16x16x4_f32(false, a1, false, b1,
                                               (short)0, c1, false, false);
  }
  const v8f c = c0 + c1;
  const int col = n0 + m;
  #pragma unroll
  for (int v = 0; v < 8; ++v)
    atomicAdd(&y_acc[(v + 8 * half) * H2 + col], c[v]);
}

// ---- K4b: g_y = W2net2 * (1 - tanh(y + b1)^2) ----------------------------
__global__ void k4b_gy(const float* __restrict__ y_acc, Params P,
                       float* __restrict__ g_y) {
  int i = blockIdx.x * 256 + threadIdx.x;  // 16*256 = 4096 exact
  int col = i & (H2 - 1);
  float z = tanhf(y_acc[i] + P.net2.b1[col]);
  g_y[i] = P.net2.W2[col] * (1.f - z * z);
}

// ---- K5 (WMMA): g_h11 = g_y @ W1net2^T  (16x10000 = 16x256 . 256x10000) --
// 625 waves; 4 independent accumulator chains over the 64 K-steps.
__global__ void k5_net2_bwd(const float* __restrict__ g_y, Params P,
                            float* __restrict__ g_h11) {
  const int lane = threadIdx.x;
  const int half = lane >> 4, m = lane & 15;
  const int n0 = blockIdx.x * 16;
  const float* __restrict__ W1 = P.net2.W1;
  const float* __restrict__ Wrow = W1 + (n0 + m) * H2;  // W1^T column strip
  v8f c0 = {}, c1 = {}, c2 = {}, c3 = {};
  for (int k = 0; k < H2; k += 16) {
    const int kk = k + 2 * half;
    v2f a0, b0, a1, b1, a2, b2, a3, b3;
    a0.x = g_y[m * H2 + kk];      a0.y = g_y[m * H2 + kk + 1];
    b0.x = Wrow[kk];              b0.y = Wrow[kk + 1];
    a1.x = g_y[m * H2 + kk + 4];  a1.y = g_y[m * H2 + kk + 5];
    b1.x = Wrow[kk + 4];          b1.y = Wrow[kk + 5];
    a2.x = g_y[m * H2 + kk + 8];  a2.y = g_y[m * H2 + kk + 9];
    b2.x = Wrow[kk + 8];          b2.y = Wrow[kk + 9];
    a3.x = g_y[m * H2 + kk + 12]; a3.y = g_y[m * H2 + kk + 13];
    b3.x = Wrow[kk + 12];         b3.y = Wrow[kk + 13];
    c0 = __builtin_amdgcn_wmma_f32_16x16x4_f32(false, a0, false, b0,
                                               (short)0, c0, false, false);
    c1 = __builtin_amdgcn_wmma_f32_16x16x4_f32(false, a1, false, b1,
                                               (short)0, c1, false, false);
    c2 = __builtin_amdgcn_wmma_f32_16x16x4_f32(false, a2, false, b2,
                                               (short)0, c2, false, false);
    c3 = __builtin_amdgcn_wmma_f32_16x16x4_f32(false, a3, false, b3,
                                               (short)0, c3, false, false);
  }
  const v8f c = (c0 + c1) + (c2 + c3);
  #pragma unroll
  for (int v = 0; v < 8; ++v)
    g_h11[(v + 8 * half) * NP + n0 + m] = c[v];
}

// ---- K6: node backward through V, V1, V3; also dT -> out[:, :NP] ---------
__global__ void k6_node_bwd(const float* __restrict__ x, Params P,
                            const float* __restrict__ agg, const float* __restrict__ vv,
                            const float* __restrict__ g_h11,
                            float* __restrict__ g_agg, float* __restrict__ g_vv,
                            float* __restrict__ g_q, float* __restrict__ out) {
  int idx = blockIdx.x * 256 + threadIdx.x;
  int b = idx / NP, n = idx % NP;
  const float* a = agg + (b * NP + n) * 5;
  float hv1[5]; mlp_hidden<5, 5>(a, P.V1.W1, P.V1.b1, hv1);
  float v1v[1]; mlp_out<5, 1>(hv1, P.V1.W2, P.V1.b2, v1v);
  float qv = x[b * 2 * NP + n];
  float hv3[5]; mlp_hidden<1, 5>(&qv, P.V3.W1, P.V3.b1, hv3);
  float v3v[1]; mlp_out<5, 1>(hv3, P.V3.W2, P.V3.b2, v3v);
  float u[3] = {v1v[0], vv[b * NP + n], v3v[0]};
  float hv[5]; mlp_hidden<3, 5>(u, P.V.W1, P.V.b1, hv);

  float g_o[1] = { g_h11[b * NP + n] };
  float g_u[3] = {0.f, 0.f, 0.f};
  mlp_bwd_in<3, 5, 1>(g_o, hv, P.V.W1, P.V.W2, g_u);

  float g_v1[1] = { g_u[0] };
  float ga[5] = {0.f, 0.f, 0.f, 0.f, 0.f};
  mlp_bwd_in<5, 5, 1>(g_v1, hv1, P.V1.W1, P.V1.W2, ga);
  float* gap = g_agg + (b * NP + n) * 5;
  #pragma unroll
  for (int i = 0; i < 5; ++i) gap[i] = ga[i];

  g_vv[b * NP + n] = g_u[1];

  float g_v3[1] = { g_u[2] };
  float gq[1] = {0.f};
  mlp_bwd_in<1, 5, 1>(g_v3, hv3, P.V3.W1, P.V3.W2, gq);
  g_q[b * NP + n] = gq[0];

  // dT: elementwise tanh-MLP gradient on momenta
  float pv = x[b * 2 * NP + NP + n];
  float dT = 0.f;
  #pragma unroll
  for (int j = 0; j < 10; ++j) {
    float t = tanhf(pv * P.T.W1[j] + P.T.b1[j]);
    dT += P.T.W2[j] * (1.f - t * t) * P.T.W1[j];
  }
  out[b * 2 * NP + n] = dT;
}

// ---- K7: edge backward A: g_node1 -> atomic scatter into g_agg -----------
__global__ void k7_edge_bwd1(const float* __restrict__ x, const int* __restrict__ ei,
                             Params P, const float* __restrict__ agg,
                             const float* __restrict__ g_vv, float* __restrict__ g_agg) {
  int idx = blockIdx.x * 256 + threadIdx.x;
  int b = idx / NE, e = idx % NE;
  int s = ei[e], r = ei[NE + e];
  const float* q = x + b * 2 * NP;
  float eij = q[s] - q[r];
  float hei[5]; mlp_hidden<1, 5>(&eij, P.edge_init.W1, P.edge_init.b1, hei);
  float ee0[5]; mlp_out<5, 5>(hei, P.edge_init.W2, P.edge_init.b2, ee0);
  const float* aS = agg + (b * NP + s) * 5;
  const float* aR = agg + (b * NP + r) * 5;
  float node1[5];
  #pragma unroll
  for (int i = 0; i < 5; ++i) node1[i] = aS[i] * aR[i];
  float h1[5]; mlp_hidden<5, 5>(node1, P.edge_up1.W1, P.edge_up1.b1, h1);
  float n11[5]; mlp_out<5, 5>(h1, P.edge_up1.W2, P.edge_up1.b2, n11);
  float h2[5]; mlp_hidden<5, 5>(node1, P.edge_up2.W1, P.edge_up2.b1, h2);
  float n12[5]; mlp_out<5, 5>(h2, P.edge_up2.W2, P.edge_up2.b2, n12);
  float ee[5];
  #pragma unroll
  for (int i = 0; i < 5; ++i) ee[i] = ee0[i] + 0.5f * (n11[i] + n12[i]);
  float hv2[5]; mlp_hidden<5, 5>(ee, P.V2.W1, P.V2.b1, hv2);

  float g_v2[1] = { g_vv[b * NP + s] };
  float g_ee[5] = {0.f, 0.f, 0.f, 0.f, 0.f};
  mlp_bwd_in<5, 5, 1>(g_v2, hv2, P.V2.W1, P.V2.W2, g_ee);
  float g_n[5];
  #pragma unroll
  for (int i = 0; i < 5; ++i) g_n[i] = 0.5f * g_ee[i];
  float g_node1[5] = {0.f, 0.f, 0.f, 0.f, 0.f};
  mlp_bwd_in<5, 5, 5>(g_n, h1, P.edge_up1.W1, P.edge_up1.W2, g_node1);
  mlp_bwd_in<5, 5, 5>(g_n, h2, P.edge_up2.W1, P.edge_up2.W2, g_node1);
  float* gR = g_agg + (b * NP + r) * 5;
  float* gS = g_agg + (b * NP + s) * 5;
  #pragma unroll
  for (int i = 0; i < 5; ++i) {
    atomicAdd(&gR[i], g_node1[i] * aS[i]);
    atomicAdd(&gS[i], g_node1[i] * aR[i]);
  }
}

// ---- K8: edge backward B: (V2 + node_up) -> edge_init -> g_q scatter -----
__global__ void k8_edge_bwd2(const float* __restrict__ x, const int* __restrict__ ei,
                             Params P, const float* __restrict__ agg,
                             const float* __restrict__ g_vv,
                             const float* __restrict__ g_agg, float* __restrict__ g_q) {
  int idx = blockIdx.x * 256 + threadIdx.x;
  int b = idx / NE, e = idx % NE;
  int s = ei[e], r = ei[NE + e];
  const float* q = x + b * 2 * NP;
  float eij = q[s] - q[r];
  float hei[5]; mlp_hidden<1, 5>(&eij, P.edge_init.W1, P.edge_init.b1, hei);
  float ee0[5]; mlp_out<5, 5>(hei, P.edge_init.W2, P.edge_init.b2, ee0);
  float hnu[5]; mlp_hidden<5, 5>(ee0, P.node_up.W1, P.node_up.b1, hnu);
  const float* aS = agg + (b * NP + s) * 5;
  const float* aR = agg + (b * NP + r) * 5;
  float node1[5];
  #pragma unroll
  for (int i = 0; i < 5; ++i) node1[i] = aS[i] * aR[i];
  float h1[5]; mlp_hidden<5, 5>(node1, P.edge_up1.W1, P.edge_up1.b1, h1);
  float n11[5]; mlp_out<5, 5>(h1, P.edge_up1.W2, P.edge_up1.b2, n11);
  float h2[5]; mlp_hidden<5, 5>(node1, P.edge_up2.W1, P.edge_up2.b1, h2);
  float n12[5]; mlp_out<5, 5>(h2, P.edge_up2.W2, P.edge_up2.b2, n12);
  float ee[5];
  #pragma unroll
  for (int i = 0; i < 5; ++i) ee[i] = ee0[i] + 0.5f * (n11[i] + n12[i]);
  float hv2[5]; mlp_hidden<5, 5>(ee, P.V2.W1, P.V2.b1, hv2);

  // d/d(ee0): from ee (identity) via V2, plus from node_up path (g_nu)
  float g_v2[1] = { g_vv[b * NP + s] };
  float g_ee0[5] = {0.f, 0.f, 0.f, 0.f, 0.f};
  mlp_bwd_in<5, 5, 1>(g_v2, hv2, P.V2.W1, P.V2.W2, g_ee0);
  float g_nu[5];
  const float* gs = g_agg + (b * NP + s) * 5;
  #pragma unroll
  for (int i = 0; i < 5; ++i) g_nu[i] = gs[i];
  mlp_bwd_in<5, 5, 5>(g_nu, hnu, P.node_up.W1, P.node_up.W2, g_ee0);

  float g_eij[1] = {0.f};
  mlp_bwd_in<1, 5, 5>(g_ee0, hei, P.edge_init.W1, P.edge_init.W2, g_eij);
  atomicAdd(&g_q[b * NP + s], g_eij[0]);
  atomicAdd(&g_q[b * NP + r], -g_eij[0]);
}

// ---- K9: out[:, NP:2NP] = -g_q -------------------------------------------
__global__ void k9_out(const float* __restrict__ g_q, float* __restrict__ out) {
  int idx = blockIdx.x * 256 + threadIdx.x;
  int b = idx / NP, n = idx % NP;
  out[b * 2 * NP + NP + n] = -g_q[b * NP + n];
}

extern "C" void kernel_launch(void* const* d_in, const int* in_sizes, int n_in,
                              void* d_out, int out_size, void* d_ws, size_t ws_size,
                              hipStream_t stream) {
  const float* x;
  const int* ei;
  Params P;
  if (n_in >= 42 && in_sizes[1] == 25) {
    // insertion-order flattening: x, params{node_up..T each W1,b1,W2,b2}, edge_index
    x = (const float*)d_in[0];
    auto M = [&](int i) {
      MLPp m; m.W1 = (const float*)d_in[i]; m.b1 = (const float*)d_in[i + 1];
      m.W2 = (const float*)d_in[i + 2]; m.b2 = (const float*)d_in[i + 3]; return m;
    };
    P.node_up = M(1); P.edge_init = M(5); P.edge_up1 = M(9); P.edge_up2 = M(13);
    P.net2 = M(17); P.V1 = M(21); P.V2 = M(25); P.V3 = M(29); P.V = M(33); P.T = M(37);
    ei = (const int*)d_in[41];
  } else {
    // jax tree_flatten sorted keys: edge_index, params{T,V,V1,V2,V3,edge_init,
    // edge_up1, edge_up2, net2, node_up each W1,W2,b1,b2}, x
    ei = (const int*)d_in[0];
    auto M = [&](int i) {
      MLPp m; m.W1 = (const float*)d_in[i]; m.W2 = (const float*)d_in[i + 1];
      m.b1 = (const float*)d_in[i + 2]; m.b2 = (const float*)d_in[i + 3]; return m;
    };
    P.T = M(1); P.V = M(5); P.V1 = M(9); P.V2 = M(13); P.V3 = M(17);
    P.edge_init = M(21); P.edge_up1 = M(25); P.edge_up2 = M(29);
    P.net2 = M(33); P.node_up = M(37);
    x = (const float*)d_in[41];
  }

  float* ws    = (float*)d_ws;
  float* agg   = ws;                 // B*NP*5 = 800000  (zeroed)
  float* vv    = agg   + 800000;     // B*NP   = 160000  (zeroed)
  float* y_acc = vv    + 160000;     // 16*256 = 4096    (zeroed)
  float* h11   = y_acc + 4096;       // 160000
  float* g_y   = h11   + 160000;     // 4096
  float* g_h11 = g_y   + 4096;       // 160000
  float* g_vv  = g_h11 + 160000;     // 160000
  float* g_agg = g_vv  + 160000;     // 800000
  float* g_q   = g_agg + 800000;     // 160000
  float* out   = (float*)d_out;

  const int ZERO_COUNT = 800000 + 160000 + 4096;  // agg + vv + y_acc
  kzero<<<(ZERO_COUNT + 255) / 256, 256, 0, stream>>>(agg, ZERO_COUNT);
  k1_edge_fwd1<<<NB * NE / 256, 256, 0, stream>>>(x, ei, P, agg);
  k2_edge_fwd2<<<NB * NE / 256, 256, 0, stream>>>(x, ei, P, agg, vv);
  k3_node_fwd<<<NB * NP / 256, 256, 0, stream>>>(x, P, agg, vv, h11);
  k4_net2_fwd<<<16 * KSPLIT, 32, 0, stream>>>(h11, P, y_acc);    // WMMA, 400 waves
  k4b_gy<<<NB, 256, 0, stream>>>(y_acc, P, g_y);
  k5_net2_bwd<<<NP / 16, 32, 0, stream>>>(g_y, P, g_h11);        // WMMA, 625 waves
  k6_node_bwd<<<NB * NP / 256, 256, 0, stream>>>(x, P, agg, vv, g_h11,
                                                 g_agg, g_vv, g_q, out);
  k7_edge_bwd1<<<NB * NE / 256, 256, 0, stream>>>(x, ei, P, agg, g_vv, g_agg);
  k8_edge_bwd2<<<NB * NE / 256, 256, 0, stream>>>(x, ei, P, agg, g_vv, g_agg, g_q);
  k9_out<<<NB * NP / 256, 256, 0, stream>>>(g_q, out);
}